// TrainableAdder_26663156973640
// MI455X (gfx1250) — compile-verified
//
#include <hip/hip_runtime.h>
#include <hip/hip_bf16.h>
#include <math.h>

// Problem constants (from reference)
#define LL     35          // sequence length
#define DD     3           // hidden dim
#define NSEQ   8           // sequences per block
#define ROWS   (NSEQ*LL)   // 280 real rows per block
#define NTHREADS 288       // 9 full wave32s (>= ROWS)
#define VOCAB  12
#define DUP    6

#define USE_ASYNC_TOKENS 1   // global_load_async_to_lds_b32 staging of tokens

typedef __attribute__((ext_vector_type(2))) float v2f;
typedef __attribute__((ext_vector_type(8))) float v8f;

#if __has_builtin(__builtin_amdgcn_wmma_f32_16x16x4_f32)
#define USE_WMMA_F32X4 1
#else
#define USE_WMMA_F32X4 0
#endif

// LDS weight blob offsets
#define W_EMBED 0    // 36
#define W_WQ    36   // 9
#define W_WK    45   // 9
#define W_WV    54   // 9
#define W_WO    63   // 9
#define W_WUP   72   // 18
#define W_BUP   90   // 6
#define W_WDN   96   // 18
#define W_WLM   114  // 36
#define W_BLM   150  // 12
#define W_TOTAL 162

__global__ __launch_bounds__(NTHREADS)
void tiny_transformer_fused(const int* __restrict__ x,
                            const float* __restrict__ embed,
                            const float* __restrict__ wq,
                            const float* __restrict__ wk,
                            const float* __restrict__ wv,
                            const float* __restrict__ wo,
                            const float* __restrict__ w_up,
                            const float* __restrict__ b_up,
                            const float* __restrict__ w_down,
                            const float* __restrict__ w_lm,
                            const float* __restrict__ b_lm,
                            float* __restrict__ out,
                            int B)
{
    // mask (phase 1/2) and output staging (phase 3) are disjoint in time -> union
    __shared__ union alignas(16) {
        float mask[3 * LL * LL];          // 14700 B
        float ob[NTHREADS * VOCAB];       // 13824 B (padded to 288 rows: no guards)
    } s_mo;
    __shared__ float s_k[NSEQ][LL][DD];   // 3360 B
    __shared__ float s_v[NSEQ][LL][DD];   // 3360 B
    __shared__ float s_hn3[NTHREADS][4];  // 4608 B (padded row: [3] = 0 -> b64 A-frag load)
    __shared__ float s_w[W_TOTAL];        // 648 B
    __shared__ int   s_tok[ROWS];         // 1120 B

    const int t  = threadIdx.x;
    const int bs = blockIdx.x * NSEQ;                   // base sequence of this block
    const int vrows = min(ROWS, B * LL - bs * LL);      // valid rows in this block

    // ---- async-stage the contiguous token chunk into LDS (CDNA5 async path) ----
#if USE_ASYNC_TOKENS
    if (t < vrows) {
        unsigned lds_off = (unsigned)(unsigned long long)&s_tok[t];   // low 32 bits = LDS offset
        unsigned long long gbase = (unsigned long long)(x + (size_t)bs * LL);
        asm volatile("global_load_async_to_lds_b32 %0, %1, %2"
                     :: "v"(lds_off), "v"(t * 4), "s"(gbase)
                     : "memory");
    }
#else
    if (t < vrows) s_tok[t] = x[(size_t)bs * LL + t];
#endif

    // ---- stage weights into LDS ----
    if (t < 36)           s_w[t] = embed[t];
    else if (t < 45)      s_w[t] = wq[t - 36];
    else if (t < 54)      s_w[t] = wk[t - 45];
    else if (t < 63)      s_w[t] = wv[t - 54];
    else if (t < 72)      s_w[t] = wo[t - 63];
    else if (t < 90)      s_w[t] = w_up[t - 72];
    else if (t < 96)      s_w[t] = b_up[t - 90];
    else if (t < 114)     s_w[t] = w_down[t - 96];
    else if (t < 150)     s_w[t] = w_lm[t - 114];
    else if (t < 162)     s_w[t] = b_lm[t - 150];

    // ---- build the fixed attention mask (translated from build_fixed_mask) ----
    const float NEG  = -1.0e30f;
    const float LN10 = 2.302585092994046f;
    for (int idx = t; idx < 3 * LL * LL; idx += NTHREADS) {
        int hd  = idx / (LL * LL);
        int rem = idx - hd * (LL * LL);
        int qq  = rem / LL;
        int j   = rem - qq * LL;
        float val = NEG;
        if (qq < 21) {
            if (j == qq) val = 0.0f;
        } else {
            int kk = qq - 21;
            if (hd == 0) {
                if (j == 10) val = 0.0f;
                else if (kk < 10 && (j == kk || j == 11 + kk)) val = 0.0f;
            } else if (hd == 1) {
                if (kk == 0) {
                    if (j == 10) val = 0.0f;
                } else {
                    int lim = kk < 10 ? kk : 10;
                    if (j < lim)                         val = -(float)(kk - j) * LN10;
                    else if (j >= 11 && (j - 11) < lim)  val = -(float)(kk - (j - 11)) * LN10;
                }
            } else { // hd == 2
                if (j == qq) val = 0.0f;
            }
        }
        s_mo.mask[idx] = val;
    }

#if USE_ASYNC_TOKENS
#if __has_builtin(__builtin_amdgcn_s_wait_asynccnt)
    __builtin_amdgcn_s_wait_asynccnt(0);
#else
    asm volatile("s_wait_asynccnt 0x0" ::: "memory");
#endif
#endif
    __syncthreads();

    // ---- phase 1: embed + rmsnorm + q/k/v ----
    float h0 = 0.f, h1 = 0.f, h2 = 0.f;
    float q0 = 0.f, q1 = 0.f, q2 = 0.f;
    int s = 0, i = 0;
    if (t < vrows) {
        s = t / LL; i = t - s * LL;
        int tok = s_tok[t];
        const float* EM = &s_w[W_EMBED];
        h0 = EM[tok * 3 + 0]; h1 = EM[tok * 3 + 1]; h2 = EM[tok * 3 + 2];
        float ms = (h0 * h0 + h1 * h1 + h2 * h2) * (1.0f / 3.0f);
        float r  = __builtin_amdgcn_rsqf(ms + 1e-6f);
        float n0 = h0 * r, n1 = h1 * r, n2 = h2 * r;
        const float* WQ = &s_w[W_WQ];
        const float* WK = &s_w[W_WK];
        const float* WV = &s_w[W_WV];
        q0 = WQ[0] * n0 + WQ[1] * n1 + WQ[2] * n2;
        q1 = WQ[3] * n0 + WQ[4] * n1 + WQ[5] * n2;
        q2 = WQ[6] * n0 + WQ[7] * n1 + WQ[8] * n2;
        s_k[s][i][0] = WK[0] * n0 + WK[1] * n1 + WK[2] * n2;
        s_k[s][i][1] = WK[3] * n0 + WK[4] * n1 + WK[5] * n2;
        s_k[s][i][2] = WK[6] * n0 + WK[7] * n1 + WK[8] * n2;
        s_v[s][i][0] = WV[0] * n0 + WV[1] * n1 + WV[2] * n2;
        s_v[s][i][1] = WV[3] * n0 + WV[4] * n1 + WV[5] * n2;
        s_v[s][i][2] = WV[6] * n0 + WV[7] * n1 + WV[8] * n2;
    }
    __syncthreads();

    // ---- phase 2: attention (head_dim==1 per head), residual, MLP, final rmsnorm ----
    if (t < vrows) {
        const float qh[3] = { q0, q1, q2 };
        float ao[3];
        #pragma unroll
        for (int hd = 0; hd < 3; ++hd) {
            const float* mrow = &s_mo.mask[hd * LL * LL + i * LL];
            float qv = qh[hd];
            float m = -3.0e38f;
            for (int j = 0; j < LL; ++j) {
                float sc = qv * s_k[s][j][hd] + mrow[j];
                m = fmaxf(m, sc);
            }
            float sum = 0.f, acc = 0.f;
            for (int j = 0; j < LL; ++j) {
                float sc = qv * s_k[s][j][hd] + mrow[j];
                float e  = __expf(sc - m);
                sum += e;
                acc += e * s_v[s][j][hd];
            }
            ao[hd] = acc / sum;
        }
        const float* WO = &s_w[W_WO];
        float g0 = h0 + WO[0] * ao[0] + WO[1] * ao[1] + WO[2] * ao[2];
        float g1 = h1 + WO[3] * ao[0] + WO[4] * ao[1] + WO[5] * ao[2];
        float g2 = h2 + WO[6] * ao[0] + WO[7] * ao[1] + WO[8] * ao[2];

        float ms2 = (g0 * g0 + g1 * g1 + g2 * g2) * (1.0f / 3.0f);
        float r2  = __builtin_amdgcn_rsqf(ms2 + 1e-6f);
        float n0 = g0 * r2, n1 = g1 * r2, n2 = g2 * r2;

        const float* WU = &s_w[W_WUP];
        const float* BU = &s_w[W_BUP];
        const float* WD = &s_w[W_WDN];
        float d0 = g0, d1 = g1, d2 = g2;
        #pragma unroll
        for (int mm = 0; mm < DUP; ++mm) {
            float u = WU[mm * 3 + 0] * n0 + WU[mm * 3 + 1] * n1 + WU[mm * 3 + 2] * n2 + BU[mm];
            u = fmaxf(u, 0.0f);
            d0 += WD[0 * DUP + mm] * u;
            d1 += WD[1 * DUP + mm] * u;
            d2 += WD[2 * DUP + mm] * u;
        }
        float ms3 = (d0 * d0 + d1 * d1 + d2 * d2) * (1.0f / 3.0f);
        float r3  = __builtin_amdgcn_rsqf(ms3 + 1e-6f);
        s_hn3[t][0] = d0 * r3;
        s_hn3[t][1] = d1 * r3;
        s_hn3[t][2] = d2 * r3;
        s_hn3[t][3] = 0.0f;
    } else {
        s_hn3[t][0] = 0.f; s_hn3[t][1] = 0.f; s_hn3[t][2] = 0.f; s_hn3[t][3] = 0.f;
    }
    __syncthreads();   // also: last use of s_mo.mask was above -> safe to reuse as s_mo.ob

    // ---- phase 3: LM head via V_WMMA_F32_16X16X4_F32, staged to LDS, coalesced writeout ----
    // D(16x16) = A(16x4: hn3 rows, K3 padded 0) * B(4x16: w_lm.T cols, padded) + C(bias rows)
#if USE_WMMA_F32X4
    {
        const int lane = t & 31;
        const int wave = t >> 5;           // wave w owns rows 32w..32w+31 = exactly 2 M-tiles
        const int col  = lane & 15;        // N index for B/C/D
        const float* WLM = &s_w[W_WLM];
        const float* BLM = &s_w[W_BLM];
        const float bias = (col < VOCAB) ? BLM[col] : 0.0f;
        const int kb = (lane < 16) ? 0 : 2;  // K pair held by this half-wave

        v2f bmat;
        bmat.x = (col < VOCAB) ? WLM[col * 3 + kb] : 0.0f;                   // K = kb
        bmat.y = (col < VOCAB && kb + 1 < 3) ? WLM[col * 3 + kb + 1] : 0.0f; // K = kb+1 (K3 pad)
        v8f c = { bias, bias, bias, bias, bias, bias, bias, bias };

        #pragma unroll
        for (int tt = 0; tt < 2; ++tt) {
            const int tile = 2 * wave + tt;
            const int arow = tile * 16 + col;      // A row held by this lane
            // branch-free A fragment: single ds_load_b64 from padded s_hn3 row
            v2f a;
            a.x = s_hn3[arow][kb];
            a.y = s_hn3[arow][kb + 1];

            // 8 args: (neg_a, A, neg_b, B, c_mod, C, reuse_a, reuse_b)
            v8f d = __builtin_amdgcn_wmma_f32_16x16x4_f32(
                false, a, false, bmat, (short)0, c, false, false);

            // stage D into LDS (padded to 288 rows -> no row guard needed)
            if (col < VOCAB) {
                const int rbase = tile * 16 + ((lane < 16) ? 0 : 8);
                #pragma unroll
                for (int r = 0; r < 8; ++r)
                    s_mo.ob[(rbase + r) * VOCAB + col] = d[r];
            }
        }
    }
#else
    // scalar fallback (kept for portability; WMMA path is the intended one)
    if (t < vrows) {
        const float* WLM = &s_w[W_WLM];
        const float* BLM = &s_w[W_BLM];
        float n0 = s_hn3[t][0], n1 = s_hn3[t][1], n2 = s_hn3[t][2];
        #pragma unroll
        for (int o = 0; o < VOCAB; ++o)
            s_mo.ob[t * VOCAB + o] =
                WLM[o * 3 + 0] * n0 + WLM[o * 3 + 1] * n1 + WLM[o * 3 + 2] * n2 + BLM[o];
    }
#endif
    __syncthreads();

    // coalesced streaming writeout: vrows*12 floats, always a multiple of 4 -> pure b128
    {
        float* gout = out + (size_t)bs * LL * VOCAB;       // 16B-aligned (bs*1680 bytes)
        const int nv4 = (vrows * VOCAB) >> 2;              // typically 840
        const float4* src = (const float4*)s_mo.ob;
        float4* dst = (float4*)gout;
        for (int idx = t; idx < nv4; idx += NTHREADS)
            dst[idx] = src[idx];
    }
}

extern "C" void kernel_launch(void* const* d_in, const int* in_sizes, int n_in,
                              void* d_out, int out_size, void* d_ws, size_t ws_size,
                              hipStream_t stream) {
    const int*   x      = (const int*)d_in[0];
    const float* embed  = (const float*)d_in[1];
    const float* wq     = (const float*)d_in[2];
    const float* wk     = (const float*)d_in[3];
    const float* wv     = (const float*)d_in[4];
    const float* wo     = (const float*)d_in[5];
    const float* w_up   = (const float*)d_in[6];
    const float* b_up   = (const float*)d_in[7];
    const float* w_down = (const float*)d_in[8];
    const float* w_lm   = (const float*)d_in[9];
    const float* b_lm   = (const float*)d_in[10];
    float* out = (float*)d_out;

    const int B = in_sizes[0] / LL;                 // 32768 sequences
    const int nblocks = (B + NSEQ - 1) / NSEQ;      // 4096 blocks

    tiny_transformer_fused<<<dim3(nblocks), dim3(NTHREADS), 0, stream>>>(
        x, embed, wq, wk, wv, wo, w_up, b_up, w_down, w_lm, b_lm, out, B);
}